// LogSignatureLayer_65798898974913
// MI455X (gfx1250) — compile-verified
//
#include <hip/hip_runtime.h>
#include <hip/hip_bf16.h>
#include <stdint.h>

typedef float v2f __attribute__((ext_vector_type(2)));
typedef float v8f __attribute__((ext_vector_type(8)));

// Problem constants (match reference)
#define NROWS 250000
#define DIM   128
#define MROWS (NROWS - 1)     // number of increments / GEMM K extent
#define GCOLS 256             // [G0 | G1] output columns
#define LSD   32897           // 1 + 256 + 256*255/2
#define KB    32              // K rows staged per block
#define LS    144             // LDS row stride in floats (bank-conflict padding)
#define NWG_MAX 120

// NOTE: no k>=M masking anywhere.  Staging clamps out-of-range rows to
// x[N-1]; the spurious tail contribution is 17*(xl (x) xl) added equally to
// the G0 and G1 halves, which cancels exactly in both B = g1rs - g1sr and
// Q = 2*g0 - g1rs - g1sr used by the finalize kernel (a stationary path has
// zero increments).

#if __has_builtin(__builtin_amdgcn_global_load_async_to_lds_b128) && \
    __has_builtin(__builtin_amdgcn_s_wait_asynccnt)
#define USE_ASYNC 1
#else
#define USE_ASYNC 0
#endif

#if USE_ASYNC
// b128 payload type per the builtin's signature (GCC-style vector of 4 ints),
// global = address_space(1), LDS = address_space(3).
typedef int b128_t __attribute__((vector_size(16)));
typedef __attribute__((address_space(1))) b128_t* gasp_t;
typedef __attribute__((address_space(3))) b128_t* lasp_t;
#endif

// ---- stage KB+1 rows of x into an LDS buffer (row index clamped) ----------
__device__ __forceinline__ void stage_block(const float* __restrict__ x,
                                            float* __restrict__ buf,
                                            int k0, int tid) {
  for (int t = tid; t < (KB + 1) * (DIM / 4); t += 256) {
    const int r  = t >> 5;
    const int c4 = (t & 31) << 2;
    int row = k0 + r;
    if (row > NROWS - 1) row = NROWS - 1;
    const float* src = x + (size_t)row * DIM + c4;
    float* dst = &buf[r * LS + c4];
#if USE_ASYNC
    // async DMA: global -> LDS, tracked by ASYNCcnt (CDNA5 path)
    __builtin_amdgcn_global_load_async_to_lds_b128(
        (gasp_t)(uintptr_t)src, (lasp_t)(uint32_t)(uintptr_t)dst, 0, 0);
#else
    const float4 v = *reinterpret_cast<const float4*>(src);
    dst[0] = v.x; dst[1] = v.y; dst[2] = v.z; dst[3] = v.w;
#endif
  }
}

__device__ __forceinline__ void wait_async_loads() {
#if USE_ASYNC
  __builtin_amdgcn_s_wait_asynccnt(0);
#endif
}

// ---- one 32-row K block of WMMA accumulation (no masking needed) ----------
__device__ __forceinline__ void compute_block(const float* __restrict__ sm,
                                              int p, int l, int h, int rshift,
                                              int cbase, v8f (&acc)[4][4]) {
#pragma unroll
  for (int kk = 0; kk < KB; kk += 4) {
    // A fragment: 16x4 fp32, element g / lane-half h holds K = g + 2h.
    const int abase = (kk + 2 * h) * LS;
    v2f av[4];
#pragma unroll
    for (int mt = 0; mt < 4; ++mt) {
      const int col = p * 64 + mt * 16 + l;
      av[mt].x = sm[abase + col];
      av[mt].y = sm[abase + LS + col];
    }
    // B fragment: 4x16 fp32, element g / lane-half h holds K row g + 2h.
    const int bbase = (kk + 2 * h + rshift) * LS;
    v2f bv[4];
#pragma unroll
    for (int nt = 0; nt < 4; ++nt) {
      const int col = cbase + nt * 16 + l;
      bv[nt].x = sm[bbase + col];
      bv[nt].y = sm[bbase + LS + col];
    }
#pragma unroll
    for (int mt = 0; mt < 4; ++mt)
#pragma unroll
      for (int nt = 0; nt < 4; ++nt)
        acc[mt][nt] = __builtin_amdgcn_wmma_f32_16x16x4_f32(
            false, av[mt], false, bv[nt], (short)0, acc[mt][nt], false, false);
  }
}

// ---------------------------------------------------------------------------
// Kernel 1: partial G = Y^T [Y | Z], double-buffered async LDS staging.
// 8 waves per WG; wave (p,q): p -> 64 M-rows, q -> 64 cols
// (q<2 -> G0 columns, q>=2 -> G1 columns via +1 row shift on B).
// ---------------------------------------------------------------------------
__global__ __launch_bounds__(256) void lsig_gram(const float* __restrict__ x,
                                                 float* __restrict__ part) {
  __shared__ float sm[2][(KB + 1) * LS];

  const int tid  = threadIdx.x;
  const int wg   = blockIdx.x;
  const int nwg  = gridDim.x;
  const int wave = tid >> 5;
  const int lane = tid & 31;
  const int p = wave >> 2;        // M half
  const int q = wave & 3;         // N quarter
  const int l = lane & 15;
  const int h = lane >> 4;        // lane half
  const int rshift   = (q >= 2) ? 1 : 0;   // +1 row for lagged (G1) columns
  const int cbase    = (q & 1) * 64;       // source column base within x row
  const int gcolbase = ((q >= 2) ? 128 : 0) + cbase;

  v8f acc[4][4];
#pragma unroll
  for (int mt = 0; mt < 4; ++mt)
#pragma unroll
    for (int nt = 0; nt < 4; ++nt)
      acc[mt][nt] = v8f{0.f, 0.f, 0.f, 0.f, 0.f, 0.f, 0.f, 0.f};

  const int numBlocks = (MROWS + KB - 1) / KB;

  int b = wg;
  if (b < numBlocks) stage_block(x, sm[0], b * KB, tid);
  wait_async_loads();
  __syncthreads();

  int cur = 0;
  for (; b < numBlocks; b += nwg) {
    const int nb = b + nwg;
    if (nb < numBlocks) stage_block(x, sm[cur ^ 1], nb * KB, tid);

    compute_block(sm[cur], p, l, h, rshift, cbase, acc);

    wait_async_loads();
    __syncthreads();
    cur ^= 1;
  }

  // ---- write partial 128x256 tile (deterministic; reduced by kernel 2) ----
  float* out = part + (size_t)wg * (DIM * GCOLS);
#pragma unroll
  for (int mt = 0; mt < 4; ++mt)
#pragma unroll
    for (int nt = 0; nt < 4; ++nt)
#pragma unroll
      for (int g = 0; g < 8; ++g) {
        const int grow = p * 64 + mt * 16 + g + 8 * h;  // D: row = g + 8h
        const int gcol = gcolbase + nt * 16 + l;
        out[(size_t)grow * GCOLS + gcol] = acc[mt][nt][g];
      }
}

// ---------------------------------------------------------------------------
// Kernel 2: deterministic reduction of per-WG partials into G (128x256).
// ---------------------------------------------------------------------------
__global__ __launch_bounds__(256) void lsig_reduce(const float* __restrict__ part,
                                                   float* __restrict__ G, int nwg) {
  const int j = blockIdx.x * 256 + threadIdx.x;  // 32768 elements
  float s = 0.f;
  for (int w = 0; w < nwg; ++w) s += part[(size_t)w * (DIM * GCOLS) + j];
  G[j] = s;
}

// ---------------------------------------------------------------------------
// Kernel 3: finalize.  B = C - C^T, Q from G0/G1 + rank-1 terms; antisym =
// [[B, B+Q], [B-Q, B]]; dot compressed upper triangle + S1 + const with W.
// One block per output o (64 blocks x 128 threads, thread = column s).
// ---------------------------------------------------------------------------
__global__ __launch_bounds__(128) void lsig_finalize(const float* __restrict__ x,
                                                     const float* __restrict__ W,
                                                     const float* __restrict__ bias,
                                                     const float* __restrict__ G,
                                                     float* __restrict__ out) {
  const int o = blockIdx.x;
  const int t = threadIdx.x;  // = s, 0..127
  __shared__ float sx0[DIM], sxl[DIM], red[DIM];
  sx0[t] = x[t];
  sxl[t] = x[(size_t)(NROWS - 1) * DIM + t];
  __syncthreads();

  const float* Wr = W + (size_t)o * LSD;
  float acc = 0.f;

  // S1 part: S1 = [delta, delta]
  const float x0s = sx0[t], xls = sxl[t];
  const float ds = xls - x0s;
  acc += (Wr[1 + t] + Wr[1 + 128 + t]) * ds;

  const int s = t;
  for (int r = 0; r < 128; ++r) {
    const float x0r = sx0[r], xlr = sxl[r];
    const float dr = xlr - x0r;
    const float g0   = G[r * GCOLS + s];
    const float g1rs = G[r * GCOLS + 128 + s];
    const float g1sr = G[s * GCOLS + 128 + r];
    const float Bv = (g1rs - g1sr) - x0r * ds + x0s * dr;
    const float Qv = 2.f * g0 - x0r * x0s + xlr * xls - g1rs - g1sr;
    const int tri = r * (511 - r) / 2;
    // uv block (a=r, e=128+s): always in the upper triangle
    acc += Wr[257 + tri + (127 + s - r)] * (Bv + Qv);
    if (s > r) {
      // uu block (a=r, e=s)
      acc += Wr[257 + tri + (s - r - 1)] * Bv;
      // vv block (a=128+r, e=128+s)
      acc += Wr[257 + (128 + r) * (383 - r) / 2 + (s - r - 1)] * Bv;
    }
  }

  red[t] = acc;
  __syncthreads();
  for (int step = 64; step > 0; step >>= 1) {
    if (t < step) red[t] += red[t + step];
    __syncthreads();
  }
  if (t == 0) out[o] = red[0] + Wr[0] + bias[o];
}

// ---------------------------------------------------------------------------
extern "C" void kernel_launch(void* const* d_in, const int* in_sizes, int n_in,
                              void* d_out, int out_size, void* d_ws, size_t ws_size,
                              hipStream_t stream) {
  const float* x = (const float*)d_in[0];
  const float* W = (const float*)d_in[1];
  const float* b = (const float*)d_in[2];
  float* out = (float*)d_out;
  float* wsf = (float*)d_ws;

  // Workspace layout (floats): [0, 32768) = G ; [32768, ...) = partials
  const size_t tileF = (size_t)DIM * GCOLS;  // 32768
  int nwg = NWG_MAX;
  const size_t availF = ws_size / sizeof(float);
  if (availF < tileF + (size_t)nwg * tileF) {
    long m = ((long)availF - (long)tileF) / (long)tileF;
    nwg = (m < 1) ? 1 : (int)m;
    if (nwg > NWG_MAX) nwg = NWG_MAX;
  }
  float* G    = wsf;
  float* part = wsf + tileF;

  lsig_gram<<<nwg, 256, 0, stream>>>(x, part);
  lsig_reduce<<<(DIM * GCOLS) / 256, 256, 0, stream>>>(part, G, nwg);
  lsig_finalize<<<64, 128, 0, stream>>>(x, W, b, G, out);
}